// EfficientMemoryGEMMWithSoftmax_88390426952068
// MI455X (gfx1250) — compile-verified
//
#include <hip/hip_runtime.h>
#include <stdint.h>

// ---------------------------------------------------------------------------
// out[b,h] = x1[b,h] (S x S softmax probs, fp32) @ x2[b,h] (S x D, fp32)
// Memory-bound on the 512MB x1 stream (AI ~31 flop/byte @ 23.3 TB/s).
// fp32 WMMA (v_wmma_f32_16x16x4_f32) for exact math; gfx1250 async
// global->LDS staging (ASYNCcnt) with double buffering so the next chunk's
// HBM traffic is in flight behind the current chunk's WMMAs.
// ---------------------------------------------------------------------------

typedef float v2f __attribute__((ext_vector_type(2)));
typedef float v8f __attribute__((ext_vector_type(8)));

#define SDIM 2048
#define DDIM 64
#define KC   32                       // K-chunk per stage
#define MT   128                      // M rows per workgroup (8 wave32 x 16)
#define NCHUNK (SDIM / KC)            // 64
#define A_STRIDE 36                   // floats/row in LDS (conflict-free b64 frags)
#define B_PAIR_STRIDE 160             // floats per (k,k+1)-pair row (halves -> disjoint banks)
#define A_BUF_FLOATS (MT * A_STRIDE)          // 4608 (18 KB)
#define B_BUF_FLOATS ((KC / 2) * B_PAIR_STRIDE) // 2560 (10 KB)  => 2 bufs = 56 KB total

__global__ __launch_bounds__(256) void gemm_probsV_wmma_async(
    const float* __restrict__ x1,   // [BH, S, S]
    const float* __restrict__ x2,   // [BH, S, D]
    float* __restrict__ out)        // [BH, S, D]
{
    __shared__ float aLds[2 * A_BUF_FLOATS];
    __shared__ float bLds[2 * B_BUF_FLOATS];

    const int tid  = threadIdx.x;
    const int lane = tid & 31;
    const int wave = tid >> 5;          // 0..7
    const int half = lane >> 4;         // K-pair select within fragment
    const int l16  = lane & 15;         // A: M row / B,C: N col

    const int m0 = blockIdx.x * MT;
    const int bh = blockIdx.y;

    const float* a_bh = x1 + (size_t)bh * SDIM * SDIM;
    const float* b_bh = x2 + (size_t)bh * SDIM * DDIM;
    float*       o_bh = out + (size_t)bh * SDIM * DDIM;

    // LDS byte-offset bases for async destinations (generic ptr low 32 bits)
    const uint32_t aBase0 = (uint32_t)(uintptr_t)(&aLds[0]);
    const uint32_t bBase0 = (uint32_t)(uintptr_t)(&bLds[0]);

    v8f acc[4];
    #pragma unroll
    for (int nt = 0; nt < 4; ++nt) {
        #pragma unroll
        for (int v = 0; v < 8; ++v) acc[nt][v] = 0.0f;
    }

    // staging coordinates
    const int aRowT  = tid >> 3;   // 0..31 ; row = aRowT + 32*i
    const int aCol4T = tid & 7;    // float4 index along K (0..7)
    const int bNT    = tid & 63;   // 0..63
    const int bKT    = tid >> 6;   // 0..3  ; k = bKT + 4*j

    // Issue one chunk's async loads: 4x b128 (A, NT-hinted stream) + 8x b32 (B,
    // scattered straight into the (k,k+1)-pair swizzle). 12 ASYNCcnt ops/thread.
    auto issueChunk = [&](int kc, int buf) {
        const uint32_t aDst = aBase0 + (uint32_t)buf * (A_BUF_FLOATS * 4);
        const uint32_t bDst = bBase0 + (uint32_t)buf * (B_BUF_FLOATS * 4);
        #pragma unroll
        for (int i = 0; i < 4; ++i) {
            const int r = aRowT + i * 32;
            const uint32_t goff = (uint32_t)(((m0 + r) * SDIM + kc + aCol4T * 4) * 4);
            const uint32_t doff = aDst + (uint32_t)((r * A_STRIDE + aCol4T * 4) * 4);
            asm volatile("global_load_async_to_lds_b128 %0, %1, %2 th:TH_LOAD_NT"
                         :: "v"(doff), "v"(goff), "s"(a_bh) : "memory");
        }
        #pragma unroll
        for (int j = 0; j < 8; ++j) {
            const int k = bKT + j * 4;
            const uint32_t goff = (uint32_t)(((kc + k) * DDIM + bNT) * 4);
            const uint32_t doff = bDst + (uint32_t)(((k >> 1) * B_PAIR_STRIDE + 2 * bNT + (k & 1)) * 4);
            asm volatile("global_load_async_to_lds_b32 %0, %1, %2"
                         :: "v"(doff), "v"(goff), "s"(b_bh) : "memory");
        }
    };

    issueChunk(0, 0);

    for (int c = 0; c < NCHUNK; ++c) {
        const int cur = c & 1;
        if (c + 1 < NCHUNK) {
            issueChunk((c + 1) * KC, 1 - cur);           // prefetch into other buffer
            asm volatile("s_wait_asynccnt 12" ::: "memory"); // chunk c done; c+1 in flight
        } else {
            asm volatile("s_wait_asynccnt 0" ::: "memory");
        }
        __syncthreads();   // all waves' chunk-c LDS writes visible

        // 8 K-steps x 4 N-tiles of v_wmma_f32_16x16x4_f32
        const float* aPtr = aLds + cur * A_BUF_FLOATS
                          + (wave * 16 + l16) * A_STRIDE + 2 * half;
        const float* bPtr = bLds + cur * B_BUF_FLOATS
                          + half * B_PAIR_STRIDE + 2 * l16;
        #pragma unroll
        for (int k0 = 0; k0 < KC; k0 += 4) {
            v2f a = *(const v2f*)(aPtr + k0);
            #pragma unroll
            for (int nt = 0; nt < 4; ++nt) {
                v2f b = *(const v2f*)(bPtr + (k0 >> 1) * B_PAIR_STRIDE + nt * 32);
                acc[nt] = __builtin_amdgcn_wmma_f32_16x16x4_f32(
                    /*neg_a=*/false, a, /*neg_b=*/false, b,
                    /*c_mod=*/(short)0, acc[nt],
                    /*reuse_a=*/false, /*reuse_b=*/false);
            }
        }
        __syncthreads();   // readers done before this buffer is overwritten
    }

    // epilogue: VGPR v, lane -> M = wave*16 + 8*half + v ; N = nt*16 + l16
    #pragma unroll
    for (int nt = 0; nt < 4; ++nt) {
        const int n = nt * 16 + l16;
        #pragma unroll
        for (int v = 0; v < 8; ++v) {
            const int m = m0 + wave * 16 + half * 8 + v;
            o_bh[(size_t)m * DDIM + n] = acc[nt][v];
        }
    }
}

extern "C" void kernel_launch(void* const* d_in, const int* in_sizes, int n_in,
                              void* d_out, int out_size, void* d_ws, size_t ws_size,
                              hipStream_t stream) {
    const float* x1 = (const float*)d_in[0];   // [2,16,2048,2048] fp32
    const float* x2 = (const float*)d_in[1];   // [2,16,2048,64]   fp32
    float* out = (float*)d_out;                // [2,16,2048,64]   fp32

    dim3 grid(SDIM / MT, 32);  // 16 M-tiles x 32 batch-heads
    dim3 block(256);           // 8 wave32s
    gemm_probsV_wmma_async<<<grid, block, 0, stream>>>(x1, x2, out);
}